// WaveletLayer_46703474377154
// MI455X (gfx1250) — compile-verified
//
#include <hip/hip_runtime.h>
#include <stdint.h>

#ifndef __has_builtin
#define __has_builtin(x) 0
#endif

#define NT  256   // 8 waves (wave32)
#define RPB 8     // rows per block -> 1024 blocks

constexpr int N0 = 4096;   // row length
constexpr int WD = 4103;   // 514 + 514 + 1026 + 2049
// LDS layout (float offsets)
constexpr int OXB0 = 0;        // x row buffer 0 (4096)
constexpr int OXB1 = 4096;     // x row buffer 1 (4096)
constexpr int OA1  = 8192;     // level-1 approx (2049)
constexpr int OA2  = 10241;    // level-2 approx (1026)
constexpr int OC   = 11267;    // concat c (4103): [a3@0, d3@514, d2@1028, d1@2054]
constexpr int OC2  = 15370;    // permuted/scaled c2 (4103)
constexpr int OR1  = 19473;    // inverse level-1 output (1026)
constexpr int OR2  = 20499;    // inverse level-2 output (2050)
constexpr int SMEM_FLOATS = 22549;   // ~90.2 KB, fits 320KB/WGP LDS

#if defined(__HIP_DEVICE_COMPILE__) && \
    __has_builtin(__builtin_amdgcn_tensor_load_to_lds) && \
    __has_builtin(__builtin_amdgcn_s_wait_tensorcnt)
#define USE_TDM 1
#else
#define USE_TDM 0
#endif

#if USE_TDM
typedef uint32_t u32x4 __attribute__((ext_vector_type(4)));
typedef int      i32x4 __attribute__((ext_vector_type(4)));
typedef int      i32x8 __attribute__((ext_vector_type(8)));

// Tensor DMA: one 4096-float row (2D tile 4096x1) global -> LDS.
// D# built per CDNA5 ISA.
__device__ __forceinline__ void tdm_load_row(const float* gsrc, void* lds_dst) {
  uint64_t ga = (uint64_t)(uintptr_t)gsrc;
  uint32_t la = (uint32_t)(uintptr_t)lds_dst;   // generic LDS ptr low 32 bits = LDS byte addr
  u32x4 g0;
  g0[0] = 1u;                                           // count=1, user mode, no gather
  g0[1] = la;                                           // lds_addr [63:32]
  g0[2] = (uint32_t)ga;                                 // global_addr[31:0]
  g0[3] = (uint32_t)((ga >> 32) & 0x01FFFFFFu)          // global_addr[56:32]
        | (2u << 30);                                   // type=2 ("image")
  i32x8 g1;
  g1[0] = (int)(2u << 16);                              // data_size=2 -> 4 bytes; mask=0
  g1[1] = (int)((uint32_t)N0 << 16);                    // tensor_dim0[15:0] @ bits 63:48
  g1[2] = (int)(1u << 16);                              // tensor_dim0 hi=0; tensor_dim1=1
  g1[3] = (int)((uint32_t)N0 << 16);                    // tile_dim0 = 4096
  g1[4] = 1;                                            // tile_dim1 = 1; tile_dim2 = 0
  g1[5] = N0;                                           // tensor_dim0_stride[31:0]
  g1[6] = 0;                                            // stride hi; tensor_dim1_stride lo
  g1[7] = 0;
  i32x4 g2 = {0, 0, 0, 0};                              // dims 2/3 unused
  i32x4 g3 = {0, 0, 0, 0};
#if __clang_major__ >= 23
  i32x8 g4 = {0, 0, 0, 0, 0, 0, 0, 0};
  __builtin_amdgcn_tensor_load_to_lds(g0, g1, g2, g3, g4, 0);
#else
  __builtin_amdgcn_tensor_load_to_lds(g0, g1, g2, g3, 0);
#endif
}
#endif  // USE_TDM

// Forward DWT level: out[t] = sum_m f[m] * s[2t+1-m], zero outside [0,n)
__device__ __forceinline__ void fwd_level(const float* __restrict__ s, int n, int nout,
    float* __restrict__ ao, float* __restrict__ dо,
    float l0, float l1, float l2, float l3,
    float h0, float h1, float h2, float h3, int tid)
{
  for (int t = tid; t < nout; t += NT) {
    const int b = 2 * t + 1;
    const float x0 = (b     < n) ? s[b]     : 0.0f;   // b >= 1 always
    const float x1 = (b - 1 < n) ? s[b - 1] : 0.0f;   // b-1 >= 0 always
    const int i2 = b - 2, i3 = b - 3;
    const float x2 = (i2 >= 0 && i2 < n) ? s[i2] : 0.0f;
    const float x3 = (i3 >= 0 && i3 < n) ? s[i3] : 0.0f;
    ao[t] = l0 * x0 + l1 * x1 + l2 * x2 + l3 * x3;
    dо[t] = h0 * x0 + h1 * x1 + h2 * x2 + h3 * x3;
  }
}

// Inverse DWT level (lhs_dilation=2, pad (1,1)): nr = 2*nd-2.
// All indices provably in [0, nd-1]; the a[:-1] trim is automatic.
__device__ __forceinline__ void inv_level(const float* __restrict__ a,
    const float* __restrict__ d, int nd, float* __restrict__ r,
    float l0, float l1, float l2, float l3,
    float h0, float h1, float h2, float h3, int tid)
{
  const int nr = 2 * nd - 2;
  for (int t = tid; t < nr; t += NT) {
    const int h = t >> 1;
    float v;
    if (t & 1) v = l1 * a[h + 1] + l3 * a[h] + h1 * d[h + 1] + h3 * d[h];
    else       v = l0 * a[h + 1] + l2 * a[h] + h0 * d[h + 1] + h2 * d[h];
    r[t] = v;
  }
}

__global__ __launch_bounds__(NT) void wavelet_layer_kernel(
    const float* __restrict__ x,
    const float* __restrict__ diag_b,
    const float* __restrict__ diag_g,
    const float* __restrict__ diag_s,
    const float* __restrict__ dec_lo,
    const float* __restrict__ dec_hi,
    const float* __restrict__ rec_lo,
    const float* __restrict__ rec_hi,
    const int*   __restrict__ perm,
    float* __restrict__ out)
{
  extern __shared__ float sm[];
  float* xbuf0 = sm + OXB0;
  float* xbuf1 = sm + OXB1;
  float* a1 = sm + OA1;
  float* a2 = sm + OA2;
  float* c  = sm + OC;
  float* c2 = sm + OC2;
  float* r1 = sm + OR1;
  float* r2 = sm + OR2;

  const int tid  = threadIdx.x;
  const int row0 = blockIdx.x * RPB;

  // Filters (uniform, L2-resident) into registers.
  const float dl0 = dec_lo[0], dl1 = dec_lo[1], dl2 = dec_lo[2], dl3 = dec_lo[3];
  const float dh0 = dec_hi[0], dh1 = dec_hi[1], dh2 = dec_hi[2], dh3 = dec_hi[3];
  const float rl0 = rec_lo[0], rl1 = rec_lo[1], rl2 = rec_lo[2], rl3 = rec_lo[3];
  const float rh0 = rec_hi[0], rh1 = rec_hi[1], rh2 = rec_hi[2], rh3 = rec_hi[3];

#if USE_TDM
  if (tid < 32) tdm_load_row(x + (size_t)row0 * N0, xbuf0);
#endif

  for (int r = 0; r < RPB; ++r) {
    const int row = row0 + r;
    float* xa = (r & 1) ? xbuf1 : xbuf0;

#if USE_TDM
    if (tid < 32) {
      if (r + 1 < RPB) {   // prefetch next row into the other buffer, then cover current
        tdm_load_row(x + (size_t)(row + 1) * N0, (r & 1) ? xbuf0 : xbuf1);
        __builtin_amdgcn_s_wait_tensorcnt(1);
      } else {
        __builtin_amdgcn_s_wait_tensorcnt(0);
      }
    }
    __syncthreads();
#else
    {
      const float4* src = (const float4*)(x + (size_t)row * N0);
      float4* dst = (float4*)xa;
      for (int i = tid; i < N0 / 4; i += NT) dst[i] = src[i];
    }
    __syncthreads();
#endif

    // a = x * diag_b (in place in LDS)
    for (int i = tid; i < N0; i += NT) xa[i] *= diag_b[i];
    __syncthreads();

    // Forward DWT: details go straight into their slot of c.
    fwd_level(xa, 4096, 2049, a1, c + 2054, dl0,dl1,dl2,dl3, dh0,dh1,dh2,dh3, tid);
    __syncthreads();
    fwd_level(a1, 2049, 1026, a2, c + 1028, dl0,dl1,dl2,dl3, dh0,dh1,dh2,dh3, tid);
    __syncthreads();
    fwd_level(a2, 1026,  514, c,  c + 514,  dl0,dl1,dl2,dl3, dh0,dh1,dh2,dh3, tid);
    __syncthreads();

    // c2[j] = c[perm[j]] * diag_g[j]
    for (int j = tid; j < WD; j += NT) c2[j] = c[perm[j]] * diag_g[j];
    __syncthreads();

    // Inverse: coeffs = c2[0:514], c2[2049:2563], c2[1026:2052], c2[514:2563]
    inv_level(c2,      c2 + 2049,  514, r1, rl0,rl1,rl2,rl3, rh0,rh1,rh2,rh3, tid);
    __syncthreads();
    inv_level(r1,      c2 + 1026, 1026, r2, rl0,rl1,rl2,rl3, rh0,rh1,rh2,rh3, tid);
    __syncthreads();

    // Final inverse level fused with diag_s scale + global store.
    {
      float* orow = out + (size_t)row * N0;
      const float* af = r2;        // len 2050, trimmed to 2049 implicitly
      const float* df = c2 + 514;  // len 2049
      for (int t = tid; t < N0; t += NT) {
        const int h = t >> 1;
        float v;
        if (t & 1) v = rl1 * af[h + 1] + rl3 * af[h] + rh1 * df[h + 1] + rh3 * df[h];
        else       v = rl0 * af[h + 1] + rl2 * af[h] + rh0 * df[h + 1] + rh2 * df[h];
        orow[t] = v * diag_s[t];
      }
    }
    __syncthreads();
  }
}

extern "C" void kernel_launch(void* const* d_in, const int* in_sizes, int n_in,
                              void* d_out, int out_size, void* d_ws, size_t ws_size,
                              hipStream_t stream) {
  (void)in_sizes; (void)n_in; (void)out_size; (void)d_ws; (void)ws_size;
  const float* x      = (const float*)d_in[0];
  const float* diag_b = (const float*)d_in[1];
  const float* diag_g = (const float*)d_in[2];
  const float* diag_s = (const float*)d_in[3];
  const float* dec_lo = (const float*)d_in[4];
  const float* dec_hi = (const float*)d_in[5];
  const float* rec_lo = (const float*)d_in[6];
  const float* rec_hi = (const float*)d_in[7];
  const int*   perm   = (const int*)d_in[8];
  float* out = (float*)d_out;

  dim3 grid(8192 / RPB), block(NT);
  size_t shmem = (size_t)SMEM_FLOATS * sizeof(float);
  hipLaunchKernelGGL(wavelet_layer_kernel, grid, block, shmem, stream,
                     x, diag_b, diag_g, diag_s, dec_lo, dec_hi, rec_lo, rec_hi, perm, out);
}